// Model_5592047420183
// MI455X (gfx1250) — compile-verified
//
#include <hip/hip_runtime.h>

typedef float v2f __attribute__((ext_vector_type(2)));
typedef float v8f __attribute__((ext_vector_type(8)));

#define IN_H   224
#define IN_W   224
#define OUT_H  12
#define OUT_W  52
#define N_IMG  4096
#define IMG_STRIDE (IN_H * IN_W)
#define ST_PITCH 226          // pad rows: 226 mod 64 banks = 34 -> conflict-free lane strides
#define BA (-0.75f)

__device__ __forceinline__ float wnear(float s) {
    // |s| <= 1: (a+2)s^3 - (a+3)s^2 + 1
    return ((BA + 2.0f) * s - (BA + 3.0f)) * s * s + 1.0f;
}
__device__ __forceinline__ float wfar(float s) {
    // 1 < s < 2: a(s^3 - 5s^2 + 8s - 4)
    return BA * (((s - 5.0f) * s + 8.0f) * s - 4.0f);
}

__global__ __launch_bounds__(256) void bicubic_resize_wmma(
    const float* __restrict__ x, float* __restrict__ out)
{
    __shared__ float sT[16][ST_PITCH];   // H-resized intermediate (rows 12..15 unused garbage)
    __shared__ float sHW[OUT_H][4];
    __shared__ int   sHI[OUT_H][4];

    const int tid = threadIdx.x;
    const int img = blockIdx.x;

    // ---- H-tap table (12 rows x 4 taps) ----
    if (tid < OUT_H) {
        const int o = tid;
        float src = (o * 223.0f) / 11.0f;
        float fi  = floorf(src);
        int   i0  = (int)fi;
        float t   = src - fi;
        sHW[o][0] = wfar(1.0f + t);  sHW[o][1] = wnear(t);
        sHW[o][2] = wnear(1.0f - t); sHW[o][3] = wfar(2.0f - t);
        sHI[o][0] = max(i0 - 1, 0);
        sHI[o][1] = min(max(i0, 0), IN_H - 1);
        sHI[o][2] = min(i0 + 1, IN_H - 1);
        sHI[o][3] = min(i0 + 2, IN_H - 1);
    }
    __syncthreads();

    // ---- stage 1: sparse H-combine; only ~46 of 224 rows streamed (NT: read-once data) ----
    const float* __restrict__ xi = x + (size_t)img * IMG_STRIDE;
    for (int t = tid; t < OUT_H * IN_W; t += 256) {
        const int o = t / IN_W;
        const int w = t - o * IN_W;
        float acc =      sHW[o][0] * __builtin_nontemporal_load(xi + sHI[o][0] * IN_W + w);
        acc = fmaf(sHW[o][1], __builtin_nontemporal_load(xi + sHI[o][1] * IN_W + w), acc);
        acc = fmaf(sHW[o][2], __builtin_nontemporal_load(xi + sHI[o][2] * IN_W + w), acc);
        acc = fmaf(sHW[o][3], __builtin_nontemporal_load(xi + sHI[o][3] * IN_W + w), acc);
        sT[o][w] = acc;
    }
    __syncthreads();

    // ---- stage 2: out(16x64) = sT(16x224) x B(224x64), B generated on the fly ----
    const int wave = tid >> 5;           // wave32
    if (wave >= 4) return;               // 4 waves = 4 N-tiles of 16
    const int lane = tid & 31;
    const int hi   = lane >> 4;          // lane half
    const int nn   = lane & 15;          // M for A-frag, N for B/D-frags
    const int p    = wave * 16 + nn;
    const int pc   = min(p, OUT_W - 1);  // clamp: D cols p>=52 are garbage, never stored

    // per-lane W-taps for output column pc (B[k][p] nonzero only at the 4 tap rows)
    float srcp = (pc * 223.0f) / 51.0f;
    float fip  = floorf(srcp);
    int   j0   = (int)fip;
    float tp   = srcp - fip;
    const float pw0 = wfar(1.0f + tp), pw1 = wnear(tp);
    const float pw2 = wnear(1.0f - tp), pw3 = wfar(2.0f - tp);
    const int jA = max(j0 - 1, 0);
    const int jB = min(max(j0, 0), IN_W - 1);
    const int jC = min(j0 + 1, IN_W - 1);
    const int jD = min(j0 + 2, IN_W - 1);

    v8f acc = {};
    for (int kk = 0; kk < IN_W / 4; ++kk) {
        const int k0 = 4 * kk + 2 * hi;  // A/B frags: VGPR0 holds K0(half0)/K2(half1), VGPR1 K1/K3
        const int k1 = k0 + 1;
        v2f a;
        a.x = sT[nn][k0];
        a.y = sT[nn][k1];
        v2f b;                            // branch-free cndmask sums (clipped duplicates add)
        b.x = ((k0 == jA) ? pw0 : 0.0f) + ((k0 == jB) ? pw1 : 0.0f)
            + ((k0 == jC) ? pw2 : 0.0f) + ((k0 == jD) ? pw3 : 0.0f);
        b.y = ((k1 == jA) ? pw0 : 0.0f) + ((k1 == jB) ? pw1 : 0.0f)
            + ((k1 == jC) ? pw2 : 0.0f) + ((k1 == jD) ? pw3 : 0.0f);
        acc = __builtin_amdgcn_wmma_f32_16x16x4_f32(
            /*neg_a=*/false, a, /*neg_b=*/false, b,
            /*c_mod=*/(short)0, acc, /*reuse_a=*/false, /*reuse_b=*/false);
    }

    // ---- fused ReLU + store (D layout: VGPR r -> M=r+8*half, lane&15 -> N) ----
    #pragma unroll
    for (int r = 0; r < 8; ++r) {
        const int M = r + 8 * hi;
        if (M < OUT_H && p < OUT_W) {
            out[(size_t)img * (OUT_H * OUT_W) + M * OUT_W + p] = fmaxf(acc[r], 0.0f);
        }
    }
}

extern "C" void kernel_launch(void* const* d_in, const int* in_sizes, int n_in,
                              void* d_out, int out_size, void* d_ws, size_t ws_size,
                              hipStream_t stream) {
    (void)in_sizes; (void)n_in; (void)out_size; (void)d_ws; (void)ws_size;
    const float* x = (const float*)d_in[0];
    float* out = (float*)d_out;
    bicubic_resize_wmma<<<N_IMG, 256, 0, stream>>>(x, out);
}